// CLAPP_layer_11467562680683
// MI455X (gfx1250) — compile-verified
//
#include <hip/hip_runtime.h>

#define N_IN  4096
#define N_HID 4096
#define BETA   0.95f
#define LR     2e-6f
#define THRESH 1.0f

typedef __attribute__((ext_vector_type(2))) float v2f;
typedef __attribute__((ext_vector_type(8))) float v8f;

__device__ __forceinline__ float wave_reduce_add(float v) {
#pragma unroll
    for (int off = 16; off > 0; off >>= 1)
        v += __shfl_xor(v, off, 32);
    return v;
}

__device__ __forceinline__ float surrogate_f(float x) {
    const float PI = 3.14159265358979323846f;
    float px = PI * x;
    return 1.0f / (PI * (1.0f + px * px));
}

// ---------------------------------------------------------------------------
// Kernel 1: cur = W_fc @ inp, then LIF step (subtract reset, thresh=1)
// one wave (32 lanes) per output row; float4 coalesced loads
// ---------------------------------------------------------------------------
__global__ void k_fc_lif(const float* __restrict__ W,
                         const float* __restrict__ x,
                         const float* __restrict__ mem,
                         float* __restrict__ spk_out,
                         float* __restrict__ mem_out,
                         float* __restrict__ cur_ws,
                         float* __restrict__ spk_ws) {
    const int wave = threadIdx.x >> 5;
    const int lane = threadIdx.x & 31;
    const int row  = blockIdx.x * 8 + wave;

    const float4* Wr = reinterpret_cast<const float4*>(W + (size_t)row * N_IN);
    const float4* xv = reinterpret_cast<const float4*>(x);

    float acc = 0.0f;
#pragma unroll 4
    for (int k = lane; k < N_IN / 4; k += 32) {
        float4 w4 = Wr[k];
        float4 x4 = xv[k];
        acc = fmaf(w4.x, x4.x, acc);
        acc = fmaf(w4.y, x4.y, acc);
        acc = fmaf(w4.z, x4.z, acc);
        acc = fmaf(w4.w, x4.w, acc);
    }
    acc = wave_reduce_add(acc);

    if (lane == 0) {
        float cur     = acc;
        float mem_new = BETA * mem[row] + cur;
        float spk     = (mem_new > THRESH) ? 1.0f : 0.0f;
        cur_ws[row]  = cur;
        spk_ws[row]  = spk;
        spk_out[row] = spk;
        mem_out[row] = mem_new - spk * THRESH;
    }
}

// ---------------------------------------------------------------------------
// Kernel 2: retro = W_retro @ spk  (rows 0..H-1)
//           feedback_new = W_pred @ spk (rows H..2H-1)
// ---------------------------------------------------------------------------
__global__ void k_dual_matvec(const float* __restrict__ W_retro,
                              const float* __restrict__ W_pred,
                              const float* __restrict__ spk,
                              float* __restrict__ retro_ws,
                              float* __restrict__ fb_out) {
    const int wave = threadIdx.x >> 5;
    const int lane = threadIdx.x & 31;
    const int r    = blockIdx.x * 8 + wave;
    const int row  = r & (N_HID - 1);
    const float* W = (r < N_HID) ? W_retro : W_pred;

    const float4* Wr = reinterpret_cast<const float4*>(W + (size_t)row * N_HID);
    const float4* xv = reinterpret_cast<const float4*>(spk);

    float acc = 0.0f;
#pragma unroll 4
    for (int k = lane; k < N_HID / 4; k += 32) {
        float4 w4 = Wr[k];
        float4 x4 = xv[k];
        acc = fmaf(w4.x, x4.x, acc);
        acc = fmaf(w4.y, x4.y, acc);
        acc = fmaf(w4.z, x4.z, acc);
        acc = fmaf(w4.w, x4.w, acc);
    }
    acc = wave_reduce_add(acc);

    if (lane == 0) {
        if (r < N_HID) retro_ws[row] = acc;
        else           fb_out[row]   = acc;
    }
}

// ---------------------------------------------------------------------------
// Kernel 3: per-element scaling vectors for the outer products
//   c = LR*bf*feedback
//   a = c * surrogate(cur)
//   b = bf * retro * surrogate(cur_prev)
// ---------------------------------------------------------------------------
__global__ void k_vectors(const float* __restrict__ feedback,
                          const float* __restrict__ cur,
                          const float* __restrict__ cur_prev,
                          const float* __restrict__ retro,
                          const float* __restrict__ bf_p,
                          float* __restrict__ a_ws,
                          float* __restrict__ b_ws,
                          float* __restrict__ c_ws) {
    const int i  = blockIdx.x * blockDim.x + threadIdx.x;
    const float bf = bf_p[0];
    float c  = LR * bf * feedback[i];
    c_ws[i]  = c;
    a_ws[i]  = c * surrogate_f(cur[i]);
    b_ws[i]  = bf * retro[i] * surrogate_f(cur_prev[i]);
}

// ---------------------------------------------------------------------------
// Kernel 4: dW_pred = outer(c, spk) via V_WMMA_F32_16X16X4_F32 (rank-1: K=0)
// one wave per 16x16 tile; streaming (non-temporal) stores
// A layout (16x4 f32): lanes 0-15 hold {K=0,K=1}; lanes 16-31 hold {K=2,K=3}
// B layout mirrors A with N; C/D: VGPR r = row (r | (lane>=16)*8), col = lane&15
// ---------------------------------------------------------------------------
__global__ void k_outer1_wmma(const float* __restrict__ rowv,   // c  [H]
                              const float* __restrict__ colv,   // spk[H]
                              float* __restrict__ out) {
    const int lane = threadIdx.x & 31;
    const int wave = threadIdx.x >> 5;
    const int tile = blockIdx.x * 8 + wave;
    const int i0 = (tile >> 8) << 4;     // 256 tiles per row of tiles
    const int j0 = (tile & 255) << 4;
    const int l15 = lane & 15;
    const bool lo = lane < 16;           // v_cndmask, no EXEC divergence at WMMA

    float av = rowv[i0 + l15];
    float bv = colv[j0 + l15];

    v2f A; A.x = lo ? av : 0.0f; A.y = 0.0f;   // A[M,0]=c[M], rest 0
    v2f B; B.x = lo ? bv : 0.0f; B.y = 0.0f;   // B[0,N]=spk[N], rest 0
    v8f C = {};
    v8f D = __builtin_amdgcn_wmma_f32_16x16x4_f32(
        /*neg_a=*/false, A, /*neg_b=*/false, B,
        /*c_mod=*/(short)0, C, /*reuse_a=*/false, /*reuse_b=*/false);

    const int col  = j0 + l15;
    const int row0 = i0 + ((lane >> 4) << 3);
    float* o = out + (size_t)row0 * N_HID + col;
#pragma unroll
    for (int r = 0; r < 8; ++r)
        __builtin_nontemporal_store(D[r], o + (size_t)r * N_HID);
}

// ---------------------------------------------------------------------------
// Kernel 5: dW_fc = outer(a, inp) + outer(b, prev_inp)  (rank-2: K=0,1)
// one WMMA computes the fused 16x16 tile exactly in fp32
// ---------------------------------------------------------------------------
__global__ void k_outer2_wmma(const float* __restrict__ a_ws,
                              const float* __restrict__ b_ws,
                              const float* __restrict__ inp,
                              const float* __restrict__ prev_inp,
                              float* __restrict__ out) {
    const int lane = threadIdx.x & 31;
    const int wave = threadIdx.x >> 5;
    const int tile = blockIdx.x * 8 + wave;
    const int i0 = (tile >> 8) << 4;
    const int j0 = (tile & 255) << 4;
    const int l15 = lane & 15;
    const bool lo = lane < 16;

    float a0 = a_ws[i0 + l15];
    float a1 = b_ws[i0 + l15];
    float b0 = inp[j0 + l15];
    float b1 = prev_inp[j0 + l15];

    v2f A; A.x = lo ? a0 : 0.0f; A.y = lo ? a1 : 0.0f;  // A[M,0]=a, A[M,1]=b
    v2f B; B.x = lo ? b0 : 0.0f; B.y = lo ? b1 : 0.0f;  // B[0,N]=inp, B[1,N]=prev_inp
    v8f C = {};
    v8f D = __builtin_amdgcn_wmma_f32_16x16x4_f32(
        false, A, false, B, (short)0, C, false, false);

    const int col  = j0 + l15;
    const int row0 = i0 + ((lane >> 4) << 3);
    float* o = out + (size_t)row0 * N_IN + col;
#pragma unroll
    for (int r = 0; r < 8; ++r)
        __builtin_nontemporal_store(D[r], o + (size_t)r * N_IN);
}

// ---------------------------------------------------------------------------
extern "C" void kernel_launch(void* const* d_in, const int* in_sizes, int n_in,
                              void* d_out, int out_size, void* d_ws, size_t ws_size,
                              hipStream_t stream) {
    const float* inp      = (const float*)d_in[0];
    const float* bf       = (const float*)d_in[1];
    const float* W_fc     = (const float*)d_in[2];
    const float* W_pred   = (const float*)d_in[3];
    const float* W_retro  = (const float*)d_in[4];
    const float* feedback = (const float*)d_in[5];
    const float* cur_prev = (const float*)d_in[6];
    const float* prev_inp = (const float*)d_in[7];
    const float* mem      = (const float*)d_in[8];

    float* out     = (float*)d_out;
    float* spk_out = out;                                   // [H]
    float* mem_out = out + N_HID;                           // [H]
    float* dwpred  = out + 2 * N_HID;                       // [H,H]
    float* dwfc    = dwpred + (size_t)N_HID * N_HID;        // [H,N]
    float* fb_out  = dwfc + (size_t)N_HID * N_IN;           // [H]

    float* ws       = (float*)d_ws;
    float* cur_ws   = ws;
    float* spk_ws   = ws + N_HID;
    float* retro_ws = ws + 2 * N_HID;
    float* a_ws     = ws + 3 * N_HID;
    float* b_ws     = ws + 4 * N_HID;
    float* c_ws     = ws + 5 * N_HID;

    // 1) cur + LIF
    k_fc_lif<<<N_HID / 8, 256, 0, stream>>>(W_fc, inp, mem,
                                            spk_out, mem_out, cur_ws, spk_ws);
    // 2) retro & feedback_new
    k_dual_matvec<<<2 * N_HID / 8, 256, 0, stream>>>(W_retro, W_pred, spk_ws,
                                                     retro_ws, fb_out);
    // 3) scaling vectors
    k_vectors<<<N_HID / 256, 256, 0, stream>>>(feedback, cur_ws, cur_prev,
                                               retro_ws, bf, a_ws, b_ws, c_ws);
    // 4) dW_pred rank-1 WMMA tiles (256x256 tiles, 8 waves/block)
    k_outer1_wmma<<<(256 * 256) / 8, 256, 0, stream>>>(c_ws, spk_ws, dwpred);
    // 5) dW_fc rank-2 WMMA tiles
    k_outer2_wmma<<<(256 * 256) / 8, 256, 0, stream>>>(a_ws, b_ws, inp, prev_inp,
                                                       dwfc);
}